// QLSTM_40450001993935
// MI455X (gfx1250) — compile-verified
//
#include <hip/hip_runtime.h>
#include <hip/hip_bf16.h>

#define NQ     4
#define DIM    16
#define TT     256
#define BB     8192
#define BLK    32      // one wave32 per workgroup: workgroup barriers are ~free
#define NDEPTH 2

typedef float v2f __attribute__((ext_vector_type(2)));
typedef float v8f __attribute__((ext_vector_type(8)));

// ---------------- persistent device scratch (recomputed every launch) -----
__device__ float g_Mre[6 * 16 * 16];   // M column-major: [set][col][out]
__device__ float g_Mim[6 * 16 * 16];
__device__ float g_Amat[8 * 32 * 32];  // real symmetric embedding of M† Z_w M

// ---------------- circuit constants --------------------------------------
__device__ __forceinline__ int cnot_map(int b, int c, int t) {
  return ((b >> (NQ - 1 - c)) & 1) ? (b ^ (1 << (NQ - 1 - t))) : b;
}

__device__ void build_perm(int* perm) {
  const int cs[8] = {0, 1, 2, 3, 0, 1, 2, 3};
  const int ts[8] = {1, 2, 3, 0, 2, 3, 0, 1};
  for (int b = 0; b < DIM; ++b) perm[b] = b;
  for (int g = 0; g < 8; ++g) {
    int tmp[DIM];
    for (int b = 0; b < DIM; ++b) tmp[b] = perm[cnot_map(b, cs[g], ts[g])];
    for (int b = 0; b < DIM; ++b) perm[b] = tmp[b];
  }
}

// ---------------- setup 1: build M (16x16 unitary of entangling layers) ---
__global__ void qlstm_setup_M(const float* Wf, const float* Wi, const float* Wu,
                              const float* Wo, const float* Wh, const float* Wy) {
  const int tid = threadIdx.x;
  if (tid >= 96) return;
  const int s   = tid >> 4;   // weight set 0..5 = {f,i,u,o,h,y}
  const int col = tid & 15;   // input basis state
  const float* Wptr[6] = {Wf, Wi, Wu, Wo, Wh, Wy};
  const float* W = Wptr[s];

  int perm[DIM];
  build_perm(perm);

  float re[DIM], im[DIM];
  for (int b = 0; b < DIM; ++b) { re[b] = (b == col) ? 1.f : 0.f; im[b] = 0.f; }

  for (int d = 0; d < NDEPTH; ++d) {
    // entangling permutation: new[b] = old[ENT_PERM[b]]
    float tr[DIM], ti[DIM];
    for (int b = 0; b < DIM; ++b) { tr[b] = re[perm[b]]; ti[b] = im[perm[b]]; }
    for (int b = 0; b < DIM; ++b) { re[b] = tr[b]; im[b] = ti[b]; }
    // Rot(phi, th, om) on each wire
    for (int q = 0; q < NQ; ++q) {
      const float phi = W[(d * NQ + q) * 3 + 0];
      const float th  = W[(d * NQ + q) * 3 + 1];
      const float om  = W[(d * NQ + q) * 3 + 2];
      float cg, sg; sincosf(0.5f * th, &sg, &cg);
      float ca, sa; sincosf(0.5f * (phi + om), &sa, &ca);
      float cb, sb; sincosf(0.5f * (phi - om), &sb, &cb);
      const float u00r =  cg * ca, u00i = -cg * sa;
      const float u01r = -sg * cb, u01i = -sg * sb;
      const float u10r =  sg * cb, u10i = -sg * sb;
      const float u11r =  cg * ca, u11i =  cg * sa;
      const int mask = 1 << (NQ - 1 - q);
      for (int b0 = 0; b0 < DIM; ++b0) {
        if (b0 & mask) continue;
        const int b1 = b0 | mask;
        const float r0 = re[b0], i0 = im[b0], r1 = re[b1], i1 = im[b1];
        re[b0] = u00r * r0 - u00i * i0 + u01r * r1 - u01i * i1;
        im[b0] = u00r * i0 + u00i * r0 + u01r * i1 + u01i * r1;
        re[b1] = u10r * r0 - u10i * i0 + u11r * r1 - u11i * i1;
        im[b1] = u10r * i0 + u10i * r0 + u11r * i1 + u11i * r1;
      }
    }
  }
  for (int out = 0; out < DIM; ++out) {
    g_Mre[(s * 16 + col) * 16 + out] = re[out];
    g_Mim[(s * 16 + col) * 16 + out] = im[out];
  }
}

// ---------------- setup 2: A_k = real embedding of H = M† Z_w M ----------
__global__ void qlstm_setup_A() {
  const int k   = blockIdx.x;           // 0..7: f,i,u,o, h0,h1,h2, y
  const int tid = threadIdx.x;          // 0..255 -> (a,b)
  const int a = tid >> 4, b = tid & 15;
  const int s_of[8] = {0, 1, 2, 3, 4, 4, 4, 5};
  const int w_of[8] = {0, 0, 0, 0, 0, 1, 2, 0};
  const int s = s_of[k], w = w_of[k];
  float hre = 0.f, him = 0.f;
  for (int out = 0; out < DIM; ++out) {
    const float sgn = ((out >> (NQ - 1 - w)) & 1) ? -1.f : 1.f;
    const float ra = g_Mre[(s * 16 + a) * 16 + out];
    const float ia = g_Mim[(s * 16 + a) * 16 + out];
    const float rb = g_Mre[(s * 16 + b) * 16 + out];
    const float ib = g_Mim[(s * 16 + b) * 16 + out];
    hre += sgn * (ra * rb + ia * ib);   // Re(conj(Ma) * Mb)
    him += sgn * (ra * ib - ia * rb);   // Im(conj(Ma) * Mb)
  }
  float* A = g_Amat + k * 32 * 32;
  A[a * 32 + b]             =  hre;     // [[Hr, -Hi],[Hi, Hr]]
  A[a * 32 + 16 + b]        = -him;
  A[(16 + a) * 32 + b]      =  him;
  A[(16 + a) * 32 + 16 + b] =  hre;
}

// ---------------- main: recurrence with WMMA quadratic forms -------------
__device__ __forceinline__ void encode_state(const float v[4], float* dst /*32 floats*/) {
  // product state: psi = ⊗_w [e*cos(a/2), conj(e)*sin(a/2)], e = exp(-i*atan(v^2)/2)
  float pr[DIM], pi[DIM];
  pr[0] = 1.f; pi[0] = 0.f;
#pragma unroll
  for (int w = 0; w < NQ; ++w) {
    const float a = atanf(v[w]);
    float ch, sh; sincosf(0.5f * a, &sh, &ch);
    const float ph = 0.5f * atanf(v[w] * v[w]);
    float cp, sp; sincosf(ph, &sp, &cp);
    const float u0r = cp * ch, u0i = -sp * ch;
    const float u1r = cp * sh, u1i =  sp * sh;
    const int len = 1 << w;
#pragma unroll
    for (int j = len - 1; j >= 0; --j) {
      const float orr = pr[j], oii = pi[j];
      pr[2 * j]     = orr * u0r - oii * u0i;
      pi[2 * j]     = orr * u0i + oii * u0r;
      pr[2 * j + 1] = orr * u1r - oii * u1i;
      pi[2 * j + 1] = orr * u1i + oii * u1r;
    }
  }
#pragma unroll
  for (int b = 0; b < DIM; ++b) { dst[b] = pr[b]; dst[16 + b] = pi[b]; }
}

__device__ __forceinline__ float quad_form(int k, int tid,
                                           const float (&Alds)[8][1024],
                                           float (&Xs)[BLK][33],
                                           float (&Ps)[BLK][33]) {
  const int lane = tid & 31;
  const int wv   = tid >> 5;
  const int n    = lane & 15;
  const int hi   = lane >> 4;       // K half-select per ISA A/B layouts
  const float* A = Alds[k];
#pragma unroll
  for (int tile = 0; tile < (BLK / 16); ++tile) {
    const int rowbase = wv * 32 + tile * 16;
#pragma unroll
    for (int n0 = 0; n0 < 32; n0 += 16) {
      v8f acc = {0.f, 0.f, 0.f, 0.f, 0.f, 0.f, 0.f, 0.f};
#pragma unroll
      for (int kk = 0; kk < 32; kk += 4) {
        v2f af, bf;
        // A-frag 16x4 f32: lanes 0-15 K={kk,kk+1}, lanes 16-31 K={kk+2,kk+3}
        af.x = Xs[rowbase + n][kk + 2 * hi];
        af.y = Xs[rowbase + n][kk + 2 * hi + 1];
        // B-frag 4x16 f32 (mirrored layout), symmetric A matrix
        bf.x = A[(kk + 2 * hi) * 32 + n0 + n];
        bf.y = A[(kk + 2 * hi + 1) * 32 + n0 + n];
        acc = __builtin_amdgcn_wmma_f32_16x16x4_f32(
            false, af, false, bf, (short)0, acc, false, false);
      }
      // D 16x16 f32: VGPR r -> row r + 8*hi, col n
#pragma unroll
      for (int r = 0; r < 8; ++r)
        Ps[rowbase + r + 8 * hi][n0 + n] = acc[r];
    }
  }
  __syncthreads();
  float q = 0.f;
#pragma unroll
  for (int j = 0; j < 32; ++j) q += Xs[tid][j] * Ps[tid][j];
  __syncthreads();
  return q;
}

__device__ __forceinline__ float sigmoidf_(float x) { return 1.f / (1.f + expf(-x)); }

__global__ __launch_bounds__(BLK) void qlstm_main(const float* __restrict__ xin,
                                                  float* __restrict__ out) {
  __shared__ float Alds[8][1024];
  __shared__ float Xs[BLK][33];
  __shared__ float Ps[BLK][33];
  const int tid  = threadIdx.x;
  const int elem = blockIdx.x * BLK + tid;

  for (int i = tid; i < 8 * 1024; i += BLK) (&Alds[0][0])[i] = g_Amat[i];
  __syncthreads();

  float ct[4] = {0.f, 0.f, 0.f, 0.f};
  float ht[3] = {0.f, 0.f, 0.f};
  const float* xrow = xin + (size_t)elem * TT;

  for (int t = 0; t < TT; ++t) {
    // ---- gate circuits: shared encoded state psi(vt) ----
    float v[4] = {xrow[t], ht[0], ht[1], ht[2]};
    encode_state(v, &Xs[tid][0]);
    __syncthreads();
    float gq[4];
    for (int k = 0; k < 4; ++k) gq[k] = quad_form(k, tid, Alds, Xs, Ps);
    const float ft = sigmoidf_(gq[0]);
    const float it = sigmoidf_(gq[1]);
    const float gt = tanhf(gq[2]);
    const float ot = sigmoidf_(gq[3]);
    float z[4];
#pragma unroll
    for (int j = 0; j < 4; ++j) {
      ct[j] = ft * ct[j] + it * gt;
      z[j]  = ot * tanhf(ct[j]);
    }
    // ---- hidden/output circuits: shared encoded state psi(z) ----
    encode_state(z, &Xs[tid][0]);
    __syncthreads();
    if (t + 1 < TT) {
      for (int k = 4; k < 7; ++k) ht[k - 4] = quad_form(k, tid, Alds, Xs, Ps);
    } else {
      out[elem] = quad_form(7, tid, Alds, Xs, Ps);  // ys[-1]
    }
  }
}

extern "C" void kernel_launch(void* const* d_in, const int* in_sizes, int n_in,
                              void* d_out, int out_size, void* d_ws, size_t ws_size,
                              hipStream_t stream) {
  (void)in_sizes; (void)n_in; (void)out_size; (void)d_ws; (void)ws_size;
  const float* xin = (const float*)d_in[0];
  const float* Wf  = (const float*)d_in[1];
  const float* Wi  = (const float*)d_in[2];
  const float* Wu  = (const float*)d_in[3];
  const float* Wo  = (const float*)d_in[4];
  const float* Wy  = (const float*)d_in[5];
  const float* Wh  = (const float*)d_in[6];
  qlstm_setup_M<<<1, 128, 0, stream>>>(Wf, Wi, Wu, Wo, Wh, Wy);
  qlstm_setup_A<<<8, 256, 0, stream>>>();
  qlstm_main<<<BB / BLK, BLK, 0, stream>>>(xin, (float*)d_out);
}